// MyIFNode_15083925144246
// MI455X (gfx1250) — compile-verified
//
#include <hip/hip_runtime.h>
#include <cstdint>
#include <cstddef>

// Integrate-and-fire scan, T=8 fused timesteps.
//   mem += x_t ; s = (mem > th) ; mem = s ? 0 : mem ; out_t = s
// Streaming, HBM-bound (~268 MB @ 23.3 TB/s ~= 11.5 us floor).
// Data path: gfx1250 async-to-LDS 4-deep ring pipeline (ASYNCcnt) for the
// 8 sequential timestep loads, NT-hinted b128 stores for the spike output.

#define T_STEPS 8
#define BLOCK   256
#define DEPTH   4                       // async pipeline depth (ring buffers)
#define SLOT_BYTES (BLOCK * 16)         // 4096 B per ring slot

// Native clang vector type: accepted by __builtin_nontemporal_* (HIP's float4
// class is not). 16-byte size/alignment -> b128 memory ops.
typedef float v4f __attribute__((ext_vector_type(4)));

// Issue one 16B async copy global->LDS for this lane's slot (NT: read-once stream).
__device__ __forceinline__ void async_load_v4(uint32_t lds_byte_off, const v4f* gptr) {
    asm volatile("global_load_async_to_lds_b128 %0, %1, off th:TH_LOAD_NT"
                 :
                 : "v"(lds_byte_off), "v"(gptr)
                 : "memory");
}

// Wait until at most n async copies are outstanding (n is compile-time const
// under full unroll, so this folds to a single s_wait_asynccnt).
__device__ __forceinline__ void wait_async(int n) {
    if      (n <= 0) asm volatile("s_wait_asynccnt 0x0" ::: "memory");
    else if (n == 1) asm volatile("s_wait_asynccnt 0x1" ::: "memory");
    else if (n == 2) asm volatile("s_wait_asynccnt 0x2" ::: "memory");
    else             asm volatile("s_wait_asynccnt 0x3" ::: "memory");
}

// One IF timestep for one element: integrate, fire (strict >), hard reset.
// Returns the spike (0.0/1.0); updates membrane in place.
__device__ __forceinline__ float if_step(float& m, float x, float th) {
    m += x;
    const bool f = m > th;          // strict: (mem - th > 0)
    m = f ? 0.0f : m;               // hard reset where spiked
    return f ? 1.0f : 0.0f;
}

__global__ __launch_bounds__(BLOCK) void if_node_kernel(
        const v4f* __restrict__ xin,         // (T * strideV4) float4-vectors
        const float* __restrict__ threshold, // scalar
        v4f* __restrict__ out,               // same shape as xin
        unsigned int strideV4)               // float4 per timestep
{
    __shared__ v4f buf[DEPTH][BLOCK];        // 16 KiB ring buffer

    const int tid = threadIdx.x;
    const size_t i = (size_t)blockIdx.x * BLOCK + tid;
    const float th = threshold[0];

    // LDS byte offset of this lane's slot 0 (low 32 bits of the flat
    // shared-aperture address == workgroup-relative LDS offset).
    const uint32_t lds_base = (uint32_t)(uintptr_t)(&buf[0][tid]);

    // Prime the pipeline: DEPTH-1 copies in flight before the loop.
#pragma unroll
    for (int t = 0; t < DEPTH - 1; ++t)
        async_load_v4(lds_base + (uint32_t)t * SLOT_BYTES,
                      xin + (size_t)t * strideV4 + i);

    float m0 = 0.0f, m1 = 0.0f, m2 = 0.0f, m3 = 0.0f;

#pragma unroll
    for (int t = 0; t < T_STEPS; ++t) {
        // Keep DEPTH copies in flight: issue t+DEPTH-1 before consuming t.
        if (t + DEPTH - 1 < T_STEPS) {
            async_load_v4(lds_base + (uint32_t)((t + DEPTH - 1) & (DEPTH - 1)) * SLOT_BYTES,
                          xin + (size_t)(t + DEPTH - 1) * strideV4 + i);
        }
        // In-order completion: slot t is ready once only the younger copies
        // (those for t+1 .. min(T-1, t+DEPTH-1)) remain outstanding.
        const int younger = (T_STEPS - 1 - t < DEPTH - 1) ? (T_STEPS - 1 - t)
                                                          : (DEPTH - 1);
        wait_async(younger);

        const v4f xi = buf[t & (DEPTH - 1)][tid];   // ds_load_b128 from own slot

        v4f s;
        s.x = if_step(m0, xi.x, th);
        s.y = if_step(m1, xi.y, th);
        s.z = if_step(m2, xi.z, th);
        s.w = if_step(m3, xi.w, th);

        // Write-once streaming output: non-temporal b128 store.
        __builtin_nontemporal_store(s, out + (size_t)t * strideV4 + i);
    }
}

extern "C" void kernel_launch(void* const* d_in, const int* in_sizes, int n_in,
                              void* d_out, int out_size, void* d_ws, size_t ws_size,
                              hipStream_t stream) {
    const v4f*   x  = (const v4f*)d_in[0];     // (256,128,32,32) f32
    const float* th = (const float*)d_in[1];   // scalar threshold
    v4f*         o  = (v4f*)d_out;

    const unsigned int totalV4  = (unsigned int)(in_sizes[0] / 4);   // 8,388,608
    const unsigned int strideV4 = totalV4 / T_STEPS;                 // 1,048,576
    const unsigned int blocks   = strideV4 / BLOCK;                  // 4096

    if_node_kernel<<<dim3(blocks), dim3(BLOCK), 0, stream>>>(x, th, o, strideV4);
}